// SpatialTransformation_3015067041812
// MI455X (gfx1250) — compile-verified
//
#include <hip/hip_runtime.h>

#ifndef __has_builtin
#define __has_builtin(x) 0
#endif

#if __has_builtin(__builtin_amdgcn_global_load_async_to_lds_b128)
#define HAVE_ASYNC_LDS 1
#else
#define HAVE_ASYNC_LDS 0
#endif

#if __has_builtin(__builtin_amdgcn_s_wait_asynccnt)
#define WAIT_ASYNC0() __builtin_amdgcn_s_wait_asynccnt(0)
#else
#define WAIT_ASYNC0() asm volatile("s_wait_asynccnt 0x0" ::: "memory")
#endif

typedef int v4i __attribute__((vector_size(16)));
typedef __attribute__((address_space(1))) v4i gv4i;
typedef __attribute__((address_space(3))) v4i lv4i;
typedef float f2 __attribute__((ext_vector_type(2)));

namespace {
constexpr int kB = 8, kO = 6, kT = 16, kL = 6, kC = 3;
constexpr int kGH = 48, kGW = 48, kHW = kGH * kGW;      // glimpse 48x48 = 2304
constexpr int kCH = 192, kCW = 192, kCHW = kCH * kCW;   // canvas 192x192 = 36864
constexpr int kTiles = 8;                               // pixel tiles per frame
constexpr int kTPB = 512;                               // 16 wave32s (4 per SIMD)
constexpr int kPixPerBlk = kCHW / kTiles;               // 4608 (9 px/thread)
constexpr int kObjStride = kC * kHW;                    // 6912 floats per object
constexpr int kYaElems = kO * kObjStride;               // 41472 floats = 162 KB LDS

__device__ __forceinline__ int iclamp(int v, int lo, int hi) {
  return v < lo ? lo : (v > hi ? hi : v);
}
} // namespace

__global__ __launch_bounds__(kTPB)
void st_compose_kernel(const float* __restrict__ y_e,
                       const float* __restrict__ y_l,
                       const float* __restrict__ Y_s,
                       const float* __restrict__ Y_a,
                       const float* __restrict__ grid,
                       float* __restrict__ out) {
  extern __shared__ float smem[];
  float* ya   = smem;                 // [kO][kC][kHW] gated appearance (162 KB)
  float* yl_s = smem + kYaElems;      // [kL][kO]
  float* ye_s = yl_s + kL * kO;       // [kO]

  const int tid  = threadIdx.x;
  const int bt   = blockIdx.x / kTiles;
  const int tile = blockIdx.x % kTiles;
  const int b = bt / kT, t = bt % kT;

  // ---- stage small per-frame tensors -------------------------------------
  if (tid < kL * kO) {
    const int l = tid / kO, o = tid % kO;
    // yl[b*T+t][l][o] = y_l[b,o,t,l]
    yl_s[tid] = y_l[(((b * kO + o) * kT) + t) * kL + l];
  }
  if (tid < kO) {
    ye_s[tid] = y_e[(b * kO + tid) * kT + t];
  }

  // ---- async DMA: raw Y_a (6 objects x 3ch x 48x48) -> LDS ---------------
#pragma unroll
  for (int o = 0; o < kO; ++o) {
    const size_t gbase = (size_t)((b * kO + o) * kT + t) * kObjStride;
    const int lbase = o * kObjStride;
    for (int j = tid * 4; j < kObjStride; j += kTPB * 4) {
      const float* gsrc = Y_a + gbase + j;
      float* ldst = ya + lbase + j;
#if HAVE_ASYNC_LDS
      __builtin_amdgcn_global_load_async_to_lds_b128(
          (gv4i*)(void*)gsrc, (lv4i*)(void*)ldst, /*offset=*/0, /*cpol=*/0);
#else
      *(float4*)ldst = *(const float4*)gsrc;
#endif
    }
  }
#if HAVE_ASYNC_LDS
  WAIT_ASYNC0();
#endif
  __syncthreads();

  // ---- gate in LDS: ya *= Y_s * y_e (one Y_s texel feeds 3 channels) -----
#pragma unroll
  for (int o = 0; o < kO; ++o) {
    const size_t sbase = (size_t)((b * kO + o) * kT + t) * kHW;
    const int lbase = o * kObjStride;
    const float ge = ye_s[o];
    for (int idx = tid; idx < kHW; idx += kTPB) {
      const float s = Y_s[sbase + idx] * ge;
      ya[lbase + idx]            *= s;
      ya[lbase + kHW + idx]      *= s;
      ya[lbase + 2 * kHW + idx]  *= s;
    }
  }
  __syncthreads();

  // ---- pull layer weights into registers ---------------------------------
  float ylr[kL][kO];
#pragma unroll
  for (int l = 0; l < kL; ++l)
#pragma unroll
    for (int o = 0; o < kO; ++o) ylr[l][o] = yl_s[l * kO + o];

  // ---- fused sample + layer-assign + composite over this pixel tile ------
  const int p0 = tile * kPixPerBlk;
  for (int p = p0 + tid; p < p0 + kPixPerBlk; p += kTPB) {
    float cov[kO];
    float ap[kO][kC];
#pragma unroll
    for (int o = 0; o < kO; ++o) {
      const size_t n = (size_t)((b * kO + o) * kT + t);
      const f2* g = (const f2*)(grid + (n * kCHW + p) * 2);
      __builtin_prefetch(g + kTPB, 0, 0);  // next pixel for this thread
      const f2 gxy = __builtin_nontemporal_load(g);  // one b64 on the hot stream
      // (coord+1)*24 - 0.5  ==  fma(coord, 24, 23.5)
      const float px = fmaf(gxy.x, 24.0f, 23.5f);
      const float py = fmaf(gxy.y, 24.0f, 23.5f);
      const float xf = floorf(px), yf = floorf(py);
      const float fx = px - xf, fy = py - yf;
      const int x0 = (int)xf, y0 = (int)yf;
      const float w00 = (1.0f - fx) * (1.0f - fy);
      const float w10 = fx * (1.0f - fy);
      const float w01 = (1.0f - fx) * fy;
      const float w11 = fx * fy;
      const int vx0 = (x0 >= 0) & (x0 < kGW);
      const int vx1 = (x0 >= -1) & (x0 < kGW - 1);
      const int vy0 = (y0 >= 0) & (y0 < kGH);
      const int vy1 = (y0 >= -1) & (y0 < kGH - 1);
      const float m00 = (vx0 & vy0) ? w00 : 0.0f;
      const float m10 = (vx1 & vy0) ? w10 : 0.0f;
      const float m01 = (vx0 & vy1) ? w01 : 0.0f;
      const float m11 = (vx1 & vy1) ? w11 : 0.0f;
      cov[o] = (m00 + m10) + (m01 + m11);
      const int xc0 = iclamp(x0, 0, kGW - 1), xc1 = iclamp(x0 + 1, 0, kGW - 1);
      const int yc0 = iclamp(y0, 0, kGH - 1), yc1 = iclamp(y0 + 1, 0, kGH - 1);
      const int i00 = yc0 * kGW + xc0, i10 = yc0 * kGW + xc1;
      const int i01 = yc1 * kGW + xc0, i11 = yc1 * kGW + xc1;
      const float* base = ya + o * kObjStride;
#pragma unroll
      for (int c = 0; c < kC; ++c) {
        const float* bc = base + c * kHW;
        ap[o][c] = fmaf(m00, bc[i00],
                   fmaf(m10, bc[i10],
                   fmaf(m01, bc[i01], m11 * bc[i11])));
      }
    }

    // layer soft-assign + clamp + back-to-front compositing
    float r0 = 0.0f, r1 = 0.0f, r2 = 0.0f;
#pragma unroll
    for (int l = 0; l < kL; ++l) {
      float s = 0.0f, a0 = 0.0f, a1 = 0.0f, a2 = 0.0f;
#pragma unroll
      for (int o = 0; o < kO; ++o) {
        const float w = ylr[l][o];
        s  = fmaf(w, cov[o],   s);
        a0 = fmaf(w, ap[o][0], a0);
        a1 = fmaf(w, ap[o][1], a1);
        a2 = fmaf(w, ap[o][2], a2);
      }
      s  = fminf(s, 1.0f);
      a0 = fminf(a0, 1.0f);
      a1 = fminf(a1, 1.0f);
      a2 = fminf(a2, 1.0f);
      const float om = 1.0f - s;
      r0 = fmaf(r0, om, a0);
      r1 = fmaf(r1, om, a1);
      r2 = fmaf(r2, om, a2);
    }

    const size_t ob = (size_t)bt * kC * kCHW + p;
    __builtin_nontemporal_store(r0, out + ob);
    __builtin_nontemporal_store(r1, out + ob + kCHW);
    __builtin_nontemporal_store(r2, out + ob + 2 * kCHW);
  }
}

extern "C" void kernel_launch(void* const* d_in, const int* in_sizes, int n_in,
                              void* d_out, int out_size, void* d_ws, size_t ws_size,
                              hipStream_t stream) {
  (void)in_sizes; (void)n_in; (void)out_size; (void)d_ws; (void)ws_size;
  const float* y_e  = (const float*)d_in[0];
  const float* y_l  = (const float*)d_in[1];
  const float* Y_s  = (const float*)d_in[2];
  const float* Y_a  = (const float*)d_in[3];
  const float* grid = (const float*)d_in[4];
  float* out = (float*)d_out;

  const size_t shmem = (size_t)(kYaElems + kL * kO + kO) * sizeof(float); // ~162 KB
  // CDNA5 WGP has 320 KB LDS; opt in to >64 KB dynamic LDS.
  (void)hipFuncSetAttribute((const void*)st_compose_kernel,
                            hipFuncAttributeMaxDynamicSharedMemorySize,
                            (int)shmem);

  dim3 grid_dim(kB * kT * kTiles);  // 1024 workgroups
  dim3 blk(kTPB);                   // 16 wave32s
  st_compose_kernel<<<grid_dim, blk, shmem, stream>>>(y_e, y_l, Y_s, Y_a, grid, out);
}